// LSTM1_76098230551035
// MI455X (gfx1250) — compile-verified
//
#include <hip/hip_runtime.h>

// ---------------------------------------------------------------------------
// Fused 2-layer LSTM (T=1024, B=16384, IN=2, H=8) + final FC, for gfx1250.
//
// One wave32 owns 16 batch elements for the whole time loop. Gate
// pre-activations: z^T = W * [h ; x]^T via V_WMMA_F32_16X16X4_F32
// (M = permuted gate rows, N = batch columns, K chunked by 4; biases ride in
// as the initial C accumulator).
//
// Gate-row permutation (chosen at weight-load time) packs, per half-wave,
// all four gates of the 4 hidden units that half owns into its visible
// C-tile rows:
//   tile0 rows = [i0,f0,i1,f1,i4,f4,i5,f5 | i2,f2,i3,f3,i6,f6,i7,f7]
//   tile1 rows = [g0,o0,g1,o1,g4,o4,g5,o5 | g2,o2,g3,o3,g6,o6,g7,o7]
// => every lane reads i,f,g,o of its unit k at constant indices z0[2k],
// z0[2k+1], z1[2k], z1[2k+1]: no cross-lane swizzles, no selects in the loop.
// The owned units {0,1,4,5} (lower) / {2,3,6,7} (upper) are exactly the
// h-values each half supplies as WMMA B-operand K-rows, so B construction is
// also select-free. Nonlinearities use gfx1250's native v_tanh_f32 when the
// builtin exists (sigmoid(x) = 0.5 + 0.5*tanh(x/2)).
// ---------------------------------------------------------------------------

typedef float v2f __attribute__((ext_vector_type(2)));
typedef float v8f __attribute__((ext_vector_type(8)));

#define HH 8  // hidden size

// dst = upper-half-lane ? tv : fv — exactly one v_cndmask, opaque to opts.
// (preamble only)
__device__ __forceinline__ float sel(float fv, float tv) {
  float r;
  unsigned m = 0xFFFF0000u;
  asm("v_cndmask_b32_e64 %0, %1, %2, %3" : "=v"(r) : "v"(fv), "v"(tv), "s"(m));
  return r;
}

// SWAPX16: lane L <-> lane L^16 (final FC reduction only)
__device__ __forceinline__ float swap16(float v) {
  return __builtin_bit_cast(
      float, __builtin_amdgcn_ds_swizzle(__builtin_bit_cast(int, v), 0x401F));
}

#if __has_builtin(__builtin_amdgcn_tanhf)
__device__ __forceinline__ float ftanh(float x) {
  return __builtin_amdgcn_tanhf(x);
}
__device__ __forceinline__ float fsigmoid(float x) {
  return __builtin_fmaf(0.5f, __builtin_amdgcn_tanhf(0.5f * x), 0.5f);
}
#else
__device__ __forceinline__ float ftanh(float x) {
  float e = __expf(-2.0f * x);
  return (1.0f - e) * __builtin_amdgcn_rcpf(1.0f + e);
}
__device__ __forceinline__ float fsigmoid(float x) {
  return __builtin_amdgcn_rcpf(1.0f + __expf(-x));
}
#endif

// D = A(16x4 f32) * B(4x16 f32) + C(16x16 f32)
__device__ __forceinline__ v8f wmma4(v2f a, v2f b, v8f c) {
  return __builtin_amdgcn_wmma_f32_16x16x4_f32(false, a, false, b, (short)0, c,
                                               false, false);
}

// Permuted tile row p (0..15) of tile tau -> original PyTorch gate row.
// p = 8*half + idx, k = idx>>1, unit u = 2*half + (k&1) + 4*(k>>1),
// second = idx&1 selects f/o vs i/g.
__device__ __forceinline__ int orig_row(int tau, int p) {
  const int half = p >> 3, idx = p & 7, k = idx >> 1, second = idx & 1;
  const int u = 2 * half + (k & 1) + 4 * (k >> 1);
  return 16 * tau + 8 * second + u;
}

// Gate math for the 4 hidden units this lane owns (slot k = unit
// 2*hf + (k&1) + 4*(k>>1)). All operands local, constant indices.
__device__ __forceinline__ void lstm_gates(v8f z0, v8f z1, float cl[4],
                                           float hl[4]) {
#pragma unroll
  for (int k = 0; k < 4; ++k) {
    float si = fsigmoid(z0[2 * k + 0]);
    float sf = fsigmoid(z0[2 * k + 1]);
    float tg = ftanh(z1[2 * k + 0]);
    float so = fsigmoid(z1[2 * k + 1]);
    float c = sf * cl[k] + si * tg;
    cl[k] = c;
    hl[k] = so * ftanh(c);
  }
}

__global__ __launch_bounds__(256) void lstm2_fused_kernel(
    const float* __restrict__ x, const float* __restrict__ wih0,
    const float* __restrict__ whh0, const float* __restrict__ bih0,
    const float* __restrict__ bhh0, const float* __restrict__ wih1,
    const float* __restrict__ whh1, const float* __restrict__ bih1,
    const float* __restrict__ bhh1, const float* __restrict__ fcw,
    const float* __restrict__ fcb, float* __restrict__ out, int T, int Bn) {
  const int lane = threadIdx.x & 31;
  const int wave = blockIdx.x * (blockDim.x >> 5) + (threadIdx.x >> 5);
  const int b0 = wave * 16;
  if (b0 >= Bn) return;  // wave-uniform: EXEC stays all-ones for WMMA

  const int r = lane & 15;         // A-row within tile / batch column (B,C)
  const int hf = (lane >> 4) & 1;  // half-wave id
  const int o2 = 2 * hf;           // K-column base for this half

  // ---- A operands (weights), permuted rows, resident for whole loop -------
  // A 16x4 layout: lane L holds row L&15; lanes 0-15 carry K=4q+0,4q+1,
  // lanes 16-31 carry K=4q+2,4q+3 (ISA 7.12.2). Row permutation only touches
  // M; K columns remain original hidden indices.
  // Layer0 logical K = [h(8) | x(2) | pad(2)] -> chunks q=0,1 hidden, q=2 in.
  v2f a0[2][3];
#pragma unroll
  for (int t = 0; t < 2; ++t) {
    const int g0 = orig_row(0, r);       // permuted row r of tile t... rows
    const int g1 = orig_row(1, r);       // differ only by +16*t:
    const int row = (t == 0) ? g0 : g1;  // orig_row(t, r)
#pragma unroll
    for (int q = 0; q < 2; ++q) {
      a0[t][q].x = whh0[(row % 32) * HH + 4 * q + o2 + 0];
      a0[t][q].y = whh0[(row % 32) * HH + 4 * q + o2 + 1];
    }
    // K=8,9 = x columns (lower lanes); K=10,11 = zero pad (upper lanes)
    a0[t][2].x = sel(wih0[(row % 32) * 2 + 0], 0.0f);
    a0[t][2].y = sel(wih0[(row % 32) * 2 + 1], 0.0f);
  }
  // Layer1 logical K = [h1(8) | h0(8)] -> chunks q=0,1 recurrent, q=2,3 in.
  v2f a1[2][4];
#pragma unroll
  for (int t = 0; t < 2; ++t) {
    const int row = orig_row(t, r);
#pragma unroll
    for (int q = 0; q < 2; ++q) {
      a1[t][q].x = whh1[row * HH + 4 * q + o2 + 0];
      a1[t][q].y = whh1[row * HH + 4 * q + o2 + 1];
      a1[t][q + 2].x = wih1[row * HH + 4 * q + o2 + 0];
      a1[t][q + 2].y = wih1[row * HH + 4 * q + o2 + 1];
    }
  }

  // ---- biases as initial C accumulators (C reg j <-> tile row j + 8*hf) ---
  v8f bz0[2], bz1[2];
#pragma unroll
  for (int t = 0; t < 2; ++t)
#pragma unroll
    for (int j = 0; j < HH; ++j) {
      const int g = orig_row(t, j + 8 * hf);
      bz0[t][j] = bih0[g] + bhh0[g];
      bz1[t][j] = bih1[g] + bhh1[g];
    }

  // ---- per-lane state: the 4 units this half owns -------------------------
  // slot k = unit o2 + (k&1) + 4*(k>>1); exactly the h values this lane
  // supplies as B-operand K-rows {o2, o2+1, 4+o2, 4+o2+1}.
  float c0l[4], h0l[4], c1l[4], h1l[4];
#pragma unroll
  for (int k = 0; k < 4; ++k) c0l[k] = h0l[k] = c1l[k] = h1l[k] = 0.0f;

  const int b = b0 + r;  // this lane's batch element

  // software-pipelined x load (x[t, b, 0:2], 8B aligned)
  v2f xv = *(const v2f*)(x + (size_t)b * 2);
  for (int t = 0; t < T; ++t) {
    const int tn = (t + 1 < T) ? (t + 1) : t;  // scalar clamp, no branch
    v2f xnx = *(const v2f*)(x + ((size_t)tn * Bn + b) * 2);

    // ---------------- layer 0: z^T = Whh0*h0^T + Wih0*x^T + bias -----------
    v8f z0 = bz0[0], z1 = bz0[1];
    {
      v2f bb;
      bb.x = h0l[0]; bb.y = h0l[1];
      z0 = wmma4(a0[0][0], bb, z0);
      z1 = wmma4(a0[1][0], bb, z1);
      bb.x = h0l[2]; bb.y = h0l[3];
      z0 = wmma4(a0[0][1], bb, z0);
      z1 = wmma4(a0[1][1], bb, z1);
      // upper-lane rows K=10,11 multiply zeroed A columns: feed xv raw.
      z0 = wmma4(a0[0][2], xv, z0);
      z1 = wmma4(a0[1][2], xv, z1);
    }
    lstm_gates(z0, z1, c0l, h0l);

    // ---------------- layer 1: z^T = Whh1*h1^T + Wih1*h0^T + bias ----------
    v8f y0 = bz1[0], y1 = bz1[1];
    {
      v2f bb;
      bb.x = h1l[0]; bb.y = h1l[1];
      y0 = wmma4(a1[0][0], bb, y0);
      y1 = wmma4(a1[1][0], bb, y1);
      bb.x = h1l[2]; bb.y = h1l[3];
      y0 = wmma4(a1[0][1], bb, y0);
      y1 = wmma4(a1[1][1], bb, y1);
      bb.x = h0l[0]; bb.y = h0l[1];
      y0 = wmma4(a1[0][2], bb, y0);
      y1 = wmma4(a1[1][2], bb, y1);
      bb.x = h0l[2]; bb.y = h0l[3];
      y0 = wmma4(a1[0][3], bb, y0);
      y1 = wmma4(a1[1][3], bb, y1);
    }
    lstm_gates(y0, y1, c1l, h1l);

    xv = xnx;
  }

  // ---- final FC: each half reduces its 4 units, combine via SWAPX16 -------
  float part = 0.0f;
#pragma unroll
  for (int k = 0; k < 4; ++k)
    part += h1l[k] * fcw[o2 + (k & 1) + 4 * (k >> 1)];
  float total = part + swap16(part) + fcb[0];
  if (hf == 0) out[b] = total;
}

extern "C" void kernel_launch(void* const* d_in, const int* in_sizes, int n_in,
                              void* d_out, int out_size, void* d_ws,
                              size_t ws_size, hipStream_t stream) {
  const float* x    = (const float*)d_in[0];
  const float* wih0 = (const float*)d_in[1];
  const float* whh0 = (const float*)d_in[2];
  const float* bih0 = (const float*)d_in[3];
  const float* bhh0 = (const float*)d_in[4];
  const float* wih1 = (const float*)d_in[5];
  const float* whh1 = (const float*)d_in[6];
  const float* bih1 = (const float*)d_in[7];
  const float* bhh1 = (const float*)d_in[8];
  const float* fcw  = (const float*)d_in[9];
  const float* fcb  = (const float*)d_in[10];
  float* out = (float*)d_out;

  const int Bn = out_size;                 // OUT = 1 -> out_size == B
  const int T  = in_sizes[0] / (2 * Bn);   // IN  = 2
  const int waves  = (Bn + 15) / 16;       // one wave per 16 batch elements
  const int blocks = (waves + 7) / 8;      // 8 waves (256 threads) per block

  hipLaunchKernelGGL(lstm2_fused_kernel, dim3(blocks), dim3(256), 0, stream, x,
                     wih0, whh0, bih0, bhh0, wih1, whh1, bih1, bhh1, fcw, fcb,
                     out, T, Bn);
}